// GCNLayer_39694087750353
// MI455X (gfx1250) — compile-verified
//
#include <hip/hip_runtime.h>

typedef __attribute__((ext_vector_type(2))) float v2f;
typedef __attribute__((ext_vector_type(8))) float v8f;

#define D 128  // D_IN == D_OUT == 128

// ---------------------------------------------------------------------------
// Phase 1: edge scatter-add.  One wave32 per edge; each lane owns 4 features.
// agg[dst] += feat[src] / out_norm[src]
// ---------------------------------------------------------------------------
__global__ __launch_bounds__(256) void gcn_scatter(
    const float* __restrict__ feat,
    const float* __restrict__ out_norm,
    const int*   __restrict__ src,
    const int*   __restrict__ dst,
    float*       __restrict__ agg,
    int E)
{
    // wave-uniform edge id (blockDim.x == 256 -> 8 waves/block)
    const int edge = blockIdx.x * 8 + (threadIdx.x >> 5);
    if (edge >= E) return;
    const int lane = threadIdx.x & 31;

    const int s = src[edge];           // wave-uniform -> scalar load
    const int d = dst[edge];
    const float inv = 1.0f / out_norm[s];

    const float4 v = *reinterpret_cast<const float4*>(feat + (size_t)s * D + lane * 4);
    float* ap = agg + (size_t)d * D + lane * 4;
    atomicAdd(ap + 0, v.x * inv);
    atomicAdd(ap + 1, v.y * inv);
    atomicAdd(ap + 2, v.z * inv);
    atomicAdd(ap + 3, v.w * inv);
}

// ---------------------------------------------------------------------------
// Phase 2: out = (agg / in_norm) @ W^T + b  via V_WMMA_F32_16X16X4_F32.
// One wave per 16x16 output tile; K=128 swept in steps of 4 (32 WMMAs).
//
// A (16x4 f32): lanes 0-15 -> M=lane, K={0,1}; lanes 16-31 -> M=lane-16, K={2,3}
// B (4x16 f32): mirrored, N=lane&15, K-pair = 2*(lane>>4)
// C/D (16x16 f32): VGPR r -> M = r + 8*(lane>>4), N = lane&15
// ---------------------------------------------------------------------------
__global__ __launch_bounds__(256) void gcn_gemm_wmma(
    const float* __restrict__ agg,
    const float* __restrict__ Wm,      // [D_OUT, D_IN] row-major
    const float* __restrict__ bias,
    const float* __restrict__ in_norm,
    float*       __restrict__ out,
    int N)
{
    const int wave = (blockIdx.x * blockDim.x + threadIdx.x) >> 5;
    const int lane = threadIdx.x & 31;

    const int rowTile = wave >> 3;           // 128/16 = 8 column tiles
    const int colTile = wave & 7;
    const int row_base = rowTile * 16;
    if (row_base >= N) return;               // wave-uniform exit
    const int n_base = colTile * 16;

    const int half = lane >> 4;              // 0: K={0,1}, 1: K={2,3}
    const int lm   = lane & 15;

    int m = row_base + lm;                   // A row (clamped for partial tiles)
    const bool m_valid = (m < N);
    if (!m_valid) m = N - 1;
    const int n = n_base + lm;               // B column == W row

    const float inv = 1.0f / in_norm[m];
    const float* arow = agg + (size_t)m * D + 2 * half;
    const float* wrow = Wm  + (size_t)n * D + 2 * half;

    v8f acc = {};
#pragma unroll
    for (int k0 = 0; k0 < D; k0 += 4) {
        v2f a = *reinterpret_cast<const v2f*>(arow + k0);
        a.x *= inv;
        a.y *= inv;
        const v2f bm = *reinterpret_cast<const v2f*>(wrow + k0);
        // 8 args: (neg_a, A, neg_b, B, c_mod, C, reuse_a, reuse_b)
        acc = __builtin_amdgcn_wmma_f32_16x16x4_f32(
            false, a, false, bm, (short)0, acc, false, false);
    }

    const float bn = bias[n];
#pragma unroll
    for (int r = 0; r < 8; ++r) {
        const int row = row_base + r + 8 * half;
        if (row < N)
            out[(size_t)row * D + n] = acc[r] + bn;
    }
}

// ---------------------------------------------------------------------------
// Launch
// ---------------------------------------------------------------------------
extern "C" void kernel_launch(void* const* d_in, const int* in_sizes, int n_in,
                              void* d_out, int out_size, void* d_ws, size_t ws_size,
                              hipStream_t stream)
{
    const float* feat     = (const float*)d_in[0];
    const float* Wm       = (const float*)d_in[1];
    const float* b        = (const float*)d_in[2];
    const float* in_norm  = (const float*)d_in[3];
    const float* out_norm = (const float*)d_in[4];
    const int*   src      = (const int*)d_in[5];
    const int*   dst      = (const int*)d_in[6];
    float*       out      = (float*)d_out;

    const int N = in_sizes[0] / D;
    const int E = in_sizes[5];

    float* agg = (float*)d_ws;               // N*D floats of scratch

    // zero the accumulator (memset node is graph-capture safe)
    hipMemsetAsync(agg, 0, (size_t)N * D * sizeof(float), stream);

    // one wave per edge, 8 waves per 256-thread block
    const int scatterBlocks = (E + 7) / 8;
    gcn_scatter<<<scatterBlocks, 256, 0, stream>>>(feat, out_norm, src, dst, agg, E);

    // one wave per 16x16 output tile
    const int rowTiles = (N + 15) / 16;
    const int waves = rowTiles * 8;
    const int gemmBlocks = (waves * 32 + 255) / 256;
    gcn_gemm_wmma<<<gemmBlocks, 256, 0, stream>>>(agg, Wm, b, in_norm, out, N);
}